// Dilation_43722767073666
// MI455X (gfx1250) — compile-verified
//
#include <hip/hip_runtime.h>
#include <stdint.h>

// 5x5x5 all-ones Conv3d ("box sum"), NCDHW (4,1,128,256,256) fp32, pad=2.
// Memory-bound: ~268 MB min traffic -> ~11.5us @ 23.3 TB/s.
// Fused separable LDS kernel:
//   - tile fill via GLOBAL_LOAD_ASYNC_TO_LDS_B32 (ASYNCcnt path, no VGPR staging)
//   - W-direction 5-tap on VALU
//   - H-direction 5-tap as banded matmul on V_WMMA_F32_16X16X4_F32
//   - D-direction 5-tap on VALU + non-temporal stores (keep L2 for halo reuse)

typedef float v2f __attribute__((ext_vector_type(2)));
typedef float v8f __attribute__((ext_vector_type(8)));

#define N_ 4
#define D_ 128
#define H_ 256
#define W_ 256

#define TD 4
#define TH 16
#define TW 32
#define PD (TD + 4)   // 8  padded depth planes
#define PH (TH + 4)   // 20 padded rows
#define PW (TW + 4)   // 36 padded cols

#if defined(__HIP_DEVICE_COMPILE__)
#if __has_builtin(__builtin_amdgcn_s_wait_asynccnt)
#define WAIT_ASYNC0() __builtin_amdgcn_s_wait_asynccnt(0)
#else
#define WAIT_ASYNC0() asm volatile("s_wait_asynccnt 0x0" ::: "memory")
#endif
#endif

__global__ __launch_bounds__(256)
void box555_kernel(const float* __restrict__ x, float* __restrict__ out) {
    // sA: raw padded input tile [PD][PH][PW]; reused later as sumHW tile [PD][TH][TW]
    __shared__ float sA[PD * PH * PW];   // 5760 floats
    // sB: w-summed tile [PD][PH][TW]
    __shared__ float sB[PD * PH * TW];   // 5120 floats

    const int tid = threadIdx.x;
    const int w0  = blockIdx.x * TW;
    const int h0  = blockIdx.y * TH;
    const int zb  = blockIdx.z;
    const int n   = zb / (D_ / TD);
    const int d0  = (zb % (D_ / TD)) * TD;

    const long nbase = (long)n * D_ * H_ * W_;
    const float* xbase = x + nbase;      // uniform per block -> SGPR pair

    // ---- tile fill: async global->LDS for in-range lanes, DS zero for halo ----
    // Disjoint LDS addresses between the two paths, so ASYNCcnt vs DScnt
    // ordering does not matter; we wait on both before the barrier.
    for (int idx = tid; idx < PD * PH * PW; idx += 256) {
        int wz = idx % PW;
        int t  = idx / PW;
        int hz = t % PH;
        int dz = t / PH;
        int gw = w0 + wz - 2;
        int gh = h0 + hz - 2;
        int gd = d0 + dz - 2;
        if ((unsigned)gw < (unsigned)W_ && (unsigned)gh < (unsigned)H_ &&
            (unsigned)gd < (unsigned)D_) {
#if defined(__HIP_DEVICE_COMPILE__)
            // Low 32 bits of the flat shared pointer == LDS byte offset (ISA 10.2).
            uint32_t lds_off = (uint32_t)(uintptr_t)&sA[idx];
            uint32_t goff    = (uint32_t)(((gd * H_ + gh) * W_ + gw) * 4);
            // GVS mode: SGPR64 base + per-lane 32-bit byte offset.
            asm volatile("global_load_async_to_lds_b32 %0, %1, %2"
                         :: "v"(lds_off), "v"(goff), "s"(xbase)
                         : "memory");
#else
            sA[idx] = xbase[(gd * H_ + gh) * W_ + gw];
#endif
        } else {
            sA[idx] = 0.0f;
        }
    }
#if defined(__HIP_DEVICE_COMPILE__)
    WAIT_ASYNC0();
#endif
    __syncthreads();

    // ---- stage A: 5-tap sliding sum along W (VALU) ----
    for (int idx = tid; idx < PD * PH * TW; idx += 256) {
        int w  = idx % TW;
        int t  = idx / TW;
        int hz = t % PH;
        int dz = t / PH;
        const float* r = &sA[(dz * PH + hz) * PW + w];
        sB[idx] = r[0] + r[1] + r[2] + r[3] + r[4];
    }
    __syncthreads();

    // ---- stage B: 5-tap sum along H as banded matmul on the matrix pipe ----
    // D(16x16) = Band(16x20) x data(20x16): out_h m sums input rows m..m+4.
    // K=20 split into 5 chunks of 4 chained through the accumulator.
    // f32 A layout (ISA 7.12.2): lane = {m=lane&15}; VGPR0 holds K=2*half+0,
    // VGPR1 holds K=2*half+1 within a K-chunk (half = lane>>4). B mirrors by
    // K-row halves; C/D: VGPR v = rows v (lanes 0-15) and v+8 (lanes 16-31).
    {
        const int wave = tid >> 5;        // 0..7 == padded depth plane dz
        const int lane = tid & 31;
        const int m    = lane & 15;       // A row / B,N column
        const int half = lane >> 4;

        for (int wc = 0; wc < TW / 16; ++wc) {
            v8f c = {};
            for (int kc = 0; kc < 5; ++kc) {
                const int k0 = 4 * kc + 2 * half;
                const int k1 = k0 + 1;
                v2f a, b;
                a.x = (k0 >= m && k0 <= m + 4) ? 1.0f : 0.0f;   // band matrix
                a.y = (k1 >= m && k1 <= m + 4) ? 1.0f : 0.0f;
                b.x = sB[(wave * PH + k0) * TW + wc * 16 + m];
                b.y = sB[(wave * PH + k1) * TW + wc * 16 + m];
#if defined(__HIP_DEVICE_COMPILE__)
                c = __builtin_amdgcn_wmma_f32_16x16x4_f32(
                        /*neg_a=*/false, a, /*neg_b=*/false, b,
                        /*c_mod=*/(short)0, c,
                        /*reuse_a=*/false, /*reuse_b=*/false);
#else
                (void)a; (void)b;
#endif
            }
            // write D back into sA, reinterpreted as [PD][TH][TW] (raw tile is dead)
            #pragma unroll
            for (int v = 0; v < 8; ++v) {
                const int row = v + half * 8;
                sA[(wave * TH + row) * TW + wc * 16 + m] = c[v];
            }
        }
    }
    __syncthreads();

    // ---- stage C: 5-tap sum along D (VALU) + non-temporal stores ----
    for (int idx = tid; idx < TD * TH * TW; idx += 256) {
        int w  = idx % TW;
        int t  = idx / TW;
        int h  = t % TH;
        int dl = t / TH;
        float s = 0.0f;
        #pragma unroll
        for (int j = 0; j < 5; ++j)
            s += sA[((dl + j) * TH + h) * TW + w];
        const long o = nbase + ((long)(d0 + dl) * H_ + (h0 + h)) * W_ + (w0 + w);
        __builtin_nontemporal_store(s, &out[o]);   // output never re-read: keep L2 for halos
    }
}

extern "C" void kernel_launch(void* const* d_in, const int* in_sizes, int n_in,
                              void* d_out, int out_size, void* d_ws, size_t ws_size,
                              hipStream_t stream) {
    (void)in_sizes; (void)n_in; (void)out_size; (void)d_ws; (void)ws_size;
    const float* x = (const float*)d_in[0];
    // d_in[1] is the all-ones 5x5x5 weight; the box-sum kernel embeds it.
    float* out = (float*)d_out;

    dim3 grid(W_ / TW, H_ / TH, (D_ / TD) * N_);   // 8 x 16 x 128
    box555_kernel<<<grid, 256, 0, stream>>>(x, out);
}